// GCN_model_10067403342284
// MI455X (gfx1250) — compile-verified
//
#include <hip/hip_runtime.h>
#include <math.h>

typedef __attribute__((ext_vector_type(2))) float v2f;
typedef __attribute__((ext_vector_type(8))) float v8f;

// ---------------- degree / normalization ----------------
__global__ void k_deg_init(float* __restrict__ deg, int n) {
  int i = blockIdx.x * blockDim.x + threadIdx.x;
  if (i < n) deg[i] = 1.0f;               // self-loop contributes 1
}

__global__ void k_deg_count(const int* __restrict__ dst, float* __restrict__ deg, int E) {
  int e = blockIdx.x * blockDim.x + threadIdx.x;
  if (e < E) atomicAdd(&deg[dst[e]], 1.0f);
}

__global__ void k_deg_rsqrt(float* __restrict__ deg, int n) {
  int i = blockIdx.x * blockDim.x + threadIdx.x;
  if (i < n) deg[i] = rsqrtf(deg[i]);     // deg >= 1 always
}

// ---------------- dense GEMM: C[M,N] = A[M,K] @ B[K,N], f32 WMMA ----------------
// One wave per 16x16 C tile. K multiple of 4 (K=128 here). M multiple of 16
// (50000 = 3125*16). N may be ragged (40): clamp+mask so EXEC stays all-1s
// for every WMMA (ISA requirement).
__global__ void k_gemm_wmma_f32(const float* __restrict__ A, const float* __restrict__ B,
                                float* __restrict__ C, int M, int K, int N) {
  const int lane = threadIdx.x & 31;
  const int wave = threadIdx.x >> 5;
  const int mt   = blockIdx.y * (blockDim.x >> 5) + wave;
  const int m0   = mt * 16;
  if (m0 >= M) return;                    // wave-uniform guard
  const int n0   = blockIdx.x * 16;
  const int hi   = lane >> 4;             // 0 for lanes 0-15, 1 for 16-31
  const int lm   = lane & 15;
  const int h2   = hi << 1;               // K sub-offset: 0 or 2
  const int ncol = n0 + lm;
  const float bmask = (ncol < N) ? 1.0f : 0.0f;
  const int   nc    = (ncol < N) ? ncol : (N - 1);   // clamped load index

  const float* arow = A + (size_t)(m0 + lm) * K;
  v8f acc = {0.f, 0.f, 0.f, 0.f, 0.f, 0.f, 0.f, 0.f};

  #pragma unroll 4
  for (int k = 0; k < K; k += 4) {
    // A 16x4 tile: lane lm row, K = k + h2 .. k + h2 + 1 (8B aligned: K=128)
    v2f a = *(const v2f*)(arow + k + h2);
    // B 4x16 tile: rows k+h2, k+h2+1, column nc
    const int kb = k + h2;
    v2f b;
    b.x = B[(size_t)kb * N + nc] * bmask;
    b.y = B[(size_t)(kb + 1) * N + nc] * bmask;
    // D = A*B + C  (8-arg form: neg_a, A, neg_b, B, c_mod, C, reuse_a, reuse_b)
    acc = __builtin_amdgcn_wmma_f32_16x16x4_f32(false, a, false, b, (short)0, acc,
                                                false, false);
  }

  if (ncol < N) {
    #pragma unroll
    for (int r = 0; r < 8; ++r) {
      const int mrow = m0 + r + (hi << 3);            // M = r (+8 for hi half)
      C[(size_t)mrow * N + ncol] = acc[r];
    }
  }
}

// ---------------- self-loop term: out[i,f] = h[i,f] * dinv[i]^2 ----------------
// Full overwrite -> also serves as the accumulator init (no memset needed).
__global__ void k_selfloop(const float* __restrict__ h, const float* __restrict__ dinv,
                           float* __restrict__ out, int n, int F) {
  size_t i = (size_t)blockIdx.x * blockDim.x + threadIdx.x;
  size_t tot = (size_t)n * F;
  if (i < tot) {
    int node = (int)(i / (size_t)F);
    float di = dinv[node];
    out[i] = h[i] * di * di;
  }
}

// ---------------- edge aggregation: out[dst] += h[src] * dinv[src]*dinv[dst] --
// One wave per edge; each lane handles 4 contiguous floats (float4 gather,
// 4x global_atomic_add_f32 scatter). h fits in L2 (25.6MB) -> gathers hit L2.
__global__ void k_edge_scatter(const float* __restrict__ h, const int* __restrict__ src,
                               const int* __restrict__ dst, const float* __restrict__ dinv,
                               float* __restrict__ out, int E, int F) {
  const int e = blockIdx.x * (blockDim.x >> 5) + (threadIdx.x >> 5);
  if (e >= E) return;
  const int lane = threadIdx.x & 31;
  const int s = src[e];
  const int d = dst[e];
  const float norm = dinv[s] * dinv[d];
  const int f = lane << 2;
  if (f < F) {
    const float4 hv = *(const float4*)(h + (size_t)s * F + f);
    float* o = out + (size_t)d * F + f;
    atomicAdd(o + 0, hv.x * norm);
    atomicAdd(o + 1, hv.y * norm);
    atomicAdd(o + 2, hv.z * norm);
    atomicAdd(o + 3, hv.w * norm);
  }
}

// ---------------- bias + ReLU ----------------
__global__ void k_bias_relu(float* __restrict__ out, const float* __restrict__ b,
                            size_t tot, int F) {
  size_t i = (size_t)blockIdx.x * blockDim.x + threadIdx.x;
  if (i < tot) {
    float v = out[i] + b[(int)(i % (size_t)F)];
    out[i] = v > 0.0f ? v : 0.0f;
  }
}

// ---------------- bias + log_softmax (one wave per row, F<=64) ----------------
__global__ void k_bias_logsoftmax(float* __restrict__ out, const float* __restrict__ b,
                                  int n, int F) {
  const int row  = blockIdx.x * (blockDim.x >> 5) + (threadIdx.x >> 5);
  const int lane = threadIdx.x & 31;
  if (row >= n) return;
  float* rp = out + (size_t)row * F;
  const int f0 = lane, f1 = lane + 32;
  float x0 = 0.f, x1 = 0.f;
  float m0 = -INFINITY, m1 = -INFINITY;
  if (f0 < F) { x0 = rp[f0] + b[f0]; m0 = x0; }
  if (f1 < F) { x1 = rp[f1] + b[f1]; m1 = x1; }
  float m = fmaxf(m0, m1);
  #pragma unroll
  for (int off = 16; off > 0; off >>= 1) m = fmaxf(m, __shfl_xor(m, off, 32));
  float s = 0.f;
  if (f0 < F) s += __expf(x0 - m);
  if (f1 < F) s += __expf(x1 - m);
  #pragma unroll
  for (int off = 16; off > 0; off >>= 1) s += __shfl_xor(s, off, 32);
  const float lse = m + __logf(s);
  if (f0 < F) rp[f0] = x0 - lse;
  if (f1 < F) rp[f1] = x1 - lse;
}

// ---------------- orchestration ----------------
extern "C" void kernel_launch(void* const* d_in, const int* in_sizes, int n_in,
                              void* d_out, int out_size, void* d_ws, size_t ws_size,
                              hipStream_t stream) {
  const float* x  = (const float*)d_in[0];
  const int*   ei = (const int*)d_in[1];       // [2, E]
  const float* W1 = (const float*)d_in[2];
  const float* b1 = (const float*)d_in[3];
  const float* W2 = (const float*)d_in[4];
  const float* b2 = (const float*)d_in[5];
  const float* W3 = (const float*)d_in[6];
  const float* b3 = (const float*)d_in[7];
  float* out = (float*)d_out;

  const int F  = 128;
  const int Nn = in_sizes[0] / F;              // 50000
  const int E  = in_sizes[1] / 2;              // 1600000
  const int Fo = in_sizes[7];                  // 40
  const int* src = ei;
  const int* dst = ei + E;

  // workspace: dinv[Nn] | hbuf[Nn*128] | abuf[Nn*128]  (~51.4 MB)
  float* dinv = (float*)d_ws;
  float* hbuf = dinv + Nn;
  float* abuf = hbuf + (size_t)Nn * F;

  const int TB = 256;
  const size_t totF  = (size_t)Nn * F;
  const size_t totFo = (size_t)Nn * Fo;

  // --- normalization coefficients ---
  k_deg_init <<<(Nn + TB - 1) / TB, TB, 0, stream>>>(dinv, Nn);
  k_deg_count<<<(E  + TB - 1) / TB, TB, 0, stream>>>(dst, dinv, E);
  k_deg_rsqrt<<<(Nn + TB - 1) / TB, TB, 0, stream>>>(dinv, Nn);

  const dim3 gemmBlk(128);                               // 4 waves / block
  const dim3 gemmGrd12(F  / 16, (Nn / 16 + 3) / 4);
  const dim3 gemmGrd3((Fo + 15) / 16, (Nn / 16 + 3) / 4);
  const int  edgeGrd = (E + 7) / 8;                      // 8 edges / 256-thr block

  // --- layer 1: h1 = x@W1; a1 = D^-1/2 A D^-1/2 h1 + b1; relu ---
  k_gemm_wmma_f32<<<gemmGrd12, gemmBlk, 0, stream>>>(x, W1, hbuf, Nn, F, F);
  k_selfloop     <<<(totF + TB - 1) / TB, TB, 0, stream>>>(hbuf, dinv, abuf, Nn, F);
  k_edge_scatter <<<edgeGrd, TB, 0, stream>>>(hbuf, src, dst, dinv, abuf, E, F);
  k_bias_relu    <<<(totF + TB - 1) / TB, TB, 0, stream>>>(abuf, b1, totF, F);

  // --- layer 2 ---
  k_gemm_wmma_f32<<<gemmGrd12, gemmBlk, 0, stream>>>(abuf, W2, hbuf, Nn, F, F);
  k_selfloop     <<<(totF + TB - 1) / TB, TB, 0, stream>>>(hbuf, dinv, abuf, Nn, F);
  k_edge_scatter <<<edgeGrd, TB, 0, stream>>>(hbuf, src, dst, dinv, abuf, E, F);
  k_bias_relu    <<<(totF + TB - 1) / TB, TB, 0, stream>>>(abuf, b2, totF, F);

  // --- layer 3 (aggregate straight into d_out) + log_softmax ---
  k_gemm_wmma_f32<<<gemmGrd3, gemmBlk, 0, stream>>>(abuf, W3, hbuf, Nn, F, Fo);
  k_selfloop     <<<(totFo + TB - 1) / TB, TB, 0, stream>>>(hbuf, dinv, out, Nn, Fo);
  k_edge_scatter <<<edgeGrd, TB, 0, stream>>>(hbuf, src, dst, dinv, out, E, Fo);
  k_bias_logsoftmax<<<(Nn + 7) / 8, TB, 0, stream>>>(out, b3, Nn, Fo);
}